// GINConvWithEdge_57672820851273
// MI455X (gfx1250) — compile-verified
//
#include <hip/hip_runtime.h>
#include <hip/hip_bf16.h>

typedef __attribute__((ext_vector_type(2))) float v2f;
typedef __attribute__((ext_vector_type(8))) float v8f;

#define NN    50000
#define EE    800000
#define DD    128
#define EDF   32
#define HH    16
#define OUTD  128
#define AGGW  144            // D + H
#define MTILES (NN / 16)     // 3125

__device__ __forceinline__ float gelu_exact(float v) {
    return 0.5f * v * (1.0f + erff(v * 0.70710678118654752f));
}

// ---------------------------------------------------------------------------
// Kernel 1: agg[N,144] = (1+eps) * x[N,128] @ ego_W[128,144]
// one wave per 16-row strip, 9 N-tiles, K=128 in steps of 4 (f32 WMMA)
// ---------------------------------------------------------------------------
__global__ void ego_gemm(const float* __restrict__ x, const float* __restrict__ egoW,
                         const float* __restrict__ eps, float* __restrict__ agg) {
    int wave = blockIdx.x * (blockDim.x >> 5) + (threadIdx.x >> 5);
    if (wave >= MTILES) return;                 // wave-uniform: EXEC stays all-1s
    int lane = threadIdx.x & 31;
    int half = lane >> 4;
    int lm   = lane & 15;
    int row0 = wave * 16;
    float alpha = 1.0f + eps[0];

    v8f acc[9] = {};
    const float* ap = x + (size_t)(row0 + lm) * DD + 2 * half;
    for (int k = 0; k < DD; k += 4) {
        v2f a; a.x = ap[k]; a.y = ap[k + 1];
        #pragma unroll
        for (int nt = 0; nt < 9; nt++) {
            const float* bp = egoW + (size_t)(k + 2 * half) * AGGW + nt * 16 + lm;
            v2f b; b.x = bp[0]; b.y = bp[AGGW];
            acc[nt] = __builtin_amdgcn_wmma_f32_16x16x4_f32(
                false, a, false, b, (short)0, acc[nt], false, false);
        }
    }
    #pragma unroll
    for (int nt = 0; nt < 9; nt++) {
        #pragma unroll
        for (int r = 0; r < 8; r++) {
            agg[(size_t)(row0 + r + 8 * half) * AGGW + nt * 16 + lm] = alpha * acc[nt][r];
        }
    }
}

// ---------------------------------------------------------------------------
// Kernel 2: per-edge: e_emb = MLP(edge_feat); atomically scatter
// [x[src] | e_emb] into agg[dst].  One wave32 per edge.
// ---------------------------------------------------------------------------
__global__ void edge_scatter(const float* __restrict__ x, const int* __restrict__ eidx,
                             const float* __restrict__ ef,
                             const float* __restrict__ eW1, const float* __restrict__ eb1,
                             const float* __restrict__ eW2, const float* __restrict__ eb2,
                             float* __restrict__ agg) {
    int lane = threadIdx.x & 31;
    long e = (long)blockIdx.x * (blockDim.x >> 5) + (threadIdx.x >> 5);
    if (e >= EE) return;
    int src = eidx[e];
    int dst = eidx[(long)EE + e];

    // edge MLP: lanes 0..15 own hidden unit j (16..31 duplicate harmlessly)
    int j = lane & 15;
    const float* f = ef + e * EDF;
    float hacc = eb1[j];
    #pragma unroll
    for (int k = 0; k < EDF; k++) hacc = fmaf(f[k], eW1[k * HH + j], hacc);
    float hid = gelu_exact(hacc);
    float emb = eb2[j];
    #pragma unroll
    for (int i = 0; i < HH; i++) emb = fmaf(__shfl(hid, i, 32), eW2[i * HH + j], emb);

    // scatter: 32 lanes x float4 == full 128-wide source row (L2-resident x)
    const float4* xr = (const float4*)(x + (size_t)src * DD);
    float4 v = xr[lane];
    float* ar = agg + (size_t)dst * AGGW;
    atomicAdd(ar + lane * 4 + 0, v.x);
    atomicAdd(ar + lane * 4 + 1, v.y);
    atomicAdd(ar + lane * 4 + 2, v.z);
    atomicAdd(ar + lane * 4 + 3, v.w);
    if (lane < HH) atomicAdd(ar + DD + lane, emb);
}

// ---------------------------------------------------------------------------
// Kernel 3: h[N,128] = agg[N,144] @ W1[144,128] + b1, plus column sum/sumsq
// ---------------------------------------------------------------------------
__global__ void h_gemm(const float* __restrict__ agg, const float* __restrict__ W1,
                       const float* __restrict__ b1, float* __restrict__ h,
                       float* __restrict__ colsum, float* __restrict__ colsumsq) {
    int wave = blockIdx.x * (blockDim.x >> 5) + (threadIdx.x >> 5);
    if (wave >= MTILES) return;
    int lane = threadIdx.x & 31;
    int half = lane >> 4;
    int lm   = lane & 15;
    int row0 = wave * 16;

    v8f acc[8] = {};
    const float* ap = agg + (size_t)(row0 + lm) * AGGW + 2 * half;
    for (int k = 0; k < AGGW; k += 4) {
        v2f a; a.x = ap[k]; a.y = ap[k + 1];
        #pragma unroll
        for (int nt = 0; nt < 8; nt++) {
            const float* bp = W1 + (size_t)(k + 2 * half) * OUTD + nt * 16 + lm;
            v2f b; b.x = bp[0]; b.y = bp[OUTD];
            acc[nt] = __builtin_amdgcn_wmma_f32_16x16x4_f32(
                false, a, false, b, (short)0, acc[nt], false, false);
        }
    }
    #pragma unroll
    for (int nt = 0; nt < 8; nt++) {
        int col = nt * 16 + lm;
        float cb = b1[col];
        float s = 0.f, s2 = 0.f;
        #pragma unroll
        for (int r = 0; r < 8; r++) {
            float v = acc[nt][r] + cb;
            h[(size_t)(row0 + r + 8 * half) * OUTD + col] = v;
            s += v; s2 += v * v;
        }
        atomicAdd(&colsum[col], s);
        atomicAdd(&colsumsq[col], s2);
    }
}

// ---------------------------------------------------------------------------
// Kernel 4: fold BN batch stats into per-column scale/shift
// ---------------------------------------------------------------------------
__global__ void bn_coeff(const float* __restrict__ colsum, const float* __restrict__ colsumsq,
                         const float* __restrict__ gamma, const float* __restrict__ beta,
                         float* __restrict__ scale, float* __restrict__ shift) {
    int c = threadIdx.x;
    if (c < OUTD) {
        const float invN = 1.0f / (float)NN;
        float mu  = colsum[c] * invN;
        float var = colsumsq[c] * invN - mu * mu;   // biased variance
        float g   = gamma[c] * rsqrtf(var + 1e-5f);
        scale[c] = g;
        shift[c] = beta[c] - mu * g;
    }
}

// ---------------------------------------------------------------------------
// Kernel 5: out = GELU(h*scale + shift) @ W2 + b2, BN+GELU fused into A-frag
// ---------------------------------------------------------------------------
__global__ void out_gemm(const float* __restrict__ h, const float* __restrict__ W2,
                         const float* __restrict__ b2, const float* __restrict__ scale,
                         const float* __restrict__ shift, float* __restrict__ out) {
    int wave = blockIdx.x * (blockDim.x >> 5) + (threadIdx.x >> 5);
    if (wave >= MTILES) return;
    int lane = threadIdx.x & 31;
    int half = lane >> 4;
    int lm   = lane & 15;
    int row0 = wave * 16;

    v8f acc[8] = {};
    const float* ap = h + (size_t)(row0 + lm) * OUTD + 2 * half;
    for (int k = 0; k < OUTD; k += 4) {
        int ka = k + 2 * half;
        v2f a;
        a.x = gelu_exact(ap[k]     * scale[ka]     + shift[ka]);
        a.y = gelu_exact(ap[k + 1] * scale[ka + 1] + shift[ka + 1]);
        #pragma unroll
        for (int nt = 0; nt < 8; nt++) {
            const float* bp = W2 + (size_t)ka * OUTD + nt * 16 + lm;
            v2f b; b.x = bp[0]; b.y = bp[OUTD];
            acc[nt] = __builtin_amdgcn_wmma_f32_16x16x4_f32(
                false, a, false, b, (short)0, acc[nt], false, false);
        }
    }
    #pragma unroll
    for (int nt = 0; nt < 8; nt++) {
        int col = nt * 16 + lm;
        float cb = b2[col];
        #pragma unroll
        for (int r = 0; r < 8; r++) {
            out[(size_t)(row0 + r + 8 * half) * OUTD + col] = acc[nt][r] + cb;
        }
    }
}

__global__ void zero_stats(float* __restrict__ p) { p[threadIdx.x] = 0.0f; }

// ---------------------------------------------------------------------------
extern "C" void kernel_launch(void* const* d_in, const int* in_sizes, int n_in,
                              void* d_out, int out_size, void* d_ws, size_t ws_size,
                              hipStream_t stream) {
    const float* x     = (const float*)d_in[0];
    const int*   eidx  = (const int*)  d_in[1];
    const float* ef    = (const float*)d_in[2];
    const float* eW1   = (const float*)d_in[3];
    const float* eb1   = (const float*)d_in[4];
    const float* eW2   = (const float*)d_in[5];
    const float* eb2   = (const float*)d_in[6];
    const float* egoW  = (const float*)d_in[7];
    const float* eps   = (const float*)d_in[8];
    const float* W1    = (const float*)d_in[9];
    const float* b1    = (const float*)d_in[10];
    const float* gamma = (const float*)d_in[11];
    const float* beta  = (const float*)d_in[12];
    const float* W2    = (const float*)d_in[13];
    const float* b2    = (const float*)d_in[14];
    float* out = (float*)d_out;

    // workspace layout (≈54.4 MB)
    float* agg      = (float*)d_ws;                       // N * 144
    float* h        = agg + (size_t)NN * AGGW;            // N * 128
    float* colsum   = h + (size_t)NN * OUTD;              // 128
    float* colsumsq = colsum + OUTD;                      // 128
    float* scale    = colsumsq + OUTD;                    // 128
    float* shift    = scale + OUTD;                       // 128

    const int gemmBlocks = (MTILES + 7) / 8;              // 8 waves / 256-thread block

    zero_stats<<<1, 256, 0, stream>>>(colsum);            // zeroes colsum+colsumsq
    ego_gemm<<<gemmBlocks, 256, 0, stream>>>(x, egoW, eps, agg);
    edge_scatter<<<EE / 8, 256, 0, stream>>>(x, eidx, ef, eW1, eb1, eW2, eb2, agg);
    h_gemm<<<gemmBlocks, 256, 0, stream>>>(agg, W1, b1, h, colsum, colsumsq);
    bn_coeff<<<1, 128, 0, stream>>>(colsum, colsumsq, gamma, beta, scale, shift);
    out_gemm<<<gemmBlocks, 256, 0, stream>>>(h, W2, b2, scale, shift, out);
}